// Subgraphing_71459665871572
// MI455X (gfx1250) — compile-verified
//
#include <hip/hip_runtime.h>
#include <stdint.h>

#define BATCH 16
#define NN    256
#define FF    128
#define KNB   32
#define INF_REF 1000000000.0f

typedef float v2f  __attribute__((ext_vector_type(2)));
typedef float v8f  __attribute__((ext_vector_type(8)));
typedef unsigned int u32x4 __attribute__((ext_vector_type(4)));
typedef int   i32x4 __attribute__((ext_vector_type(4)));
typedef int   i32x8 __attribute__((ext_vector_type(8)));

// ---------------------------------------------------------------------------
// K1: D = where(adj>0, adj, INF); diag = 0
// ---------------------------------------------------------------------------
__global__ void __launch_bounds__(256) k_init(const float* __restrict__ adj,
                                              float* __restrict__ D) {
    int idx = blockIdx.x * 256 + threadIdx.x;      // < BATCH*NN*NN
    int ij  = idx & (NN * NN - 1);
    int i   = ij >> 8;
    int j   = ij & (NN - 1);
    float a = adj[idx];
    float d = (a > 0.0f) ? a : INF_REF;
    if (i == j) d = 0.0f;
    D[idx] = d;
}

// ---------------------------------------------------------------------------
// K2: one min-plus squaring step. grid = BATCH*4 blocks; block computes a
// 64x256 tile. The 64x256 A-panel (64KB) is DMA'd into LDS by the Tensor
// Data Mover; B values stream from L2 (whole D per batch is 256KB).
// ---------------------------------------------------------------------------
__global__ void __launch_bounds__(256) k_minplus(const float* __restrict__ Din,
                                                 float* __restrict__ Dout) {
    __shared__ float As[64 * NN];                  // 64 KB
    int b  = blockIdx.x >> 2;
    int rb = (blockIdx.x & 3) * 64;
    const float* Db = Din + b * NN * NN;

    if (threadIdx.x < 32) {                        // wave 0 issues the TDM op
        uint32_t lds = (uint32_t)(uintptr_t)(&As[0]);
        uint64_t ga  = (uint64_t)(uintptr_t)(Db + rb * NN);
        u32x4 g0; i32x8 g1; i32x4 g2; i32x4 g3; i32x8 g4;
        // group0: count=1 | lds_addr | global_addr | type=2
        g0[0] = 1u;
        g0[1] = lds;
        g0[2] = (uint32_t)ga;
        g0[3] = (uint32_t)((ga >> 32) & 0x01FFFFFFu) | (2u << 30);
        // group1: data_size=4B(code2); tensor_dim0=256, tensor_dim1=256;
        //         tile_dim0=256, tile_dim1=64; tensor_dim0_stride=256
        g1[0] = (int)(2u << 16);                   // data_size
        g1[1] = (int)(256u << 16);                 // tensor_dim0[15:0] @bits63:48
        g1[2] = (int)(256u << 16);                 // tensor_dim1[15:0] @bits95:80
        g1[3] = (int)(256u << 16);                 // tile_dim0 @bits127:112
        g1[4] = 64;                                // tile_dim1 @bits143:128
        g1[5] = 256;                               // tensor_dim0_stride lo32
        g1[6] = 0;
        g1[7] = 0;
        g2 = (i32x4){0, 0, 0, 0};                  // 2D tensor: groups 2/3 unused
        g3 = (i32x4){0, 0, 0, 0};
        g4 = (i32x8){0, 0, 0, 0, 0, 0, 0, 0};
        __builtin_amdgcn_tensor_load_to_lds(g0, g1, g2, g3, g4, 0);
        __builtin_amdgcn_s_wait_tensorcnt(0);
    }
    __syncthreads();

    int tc = threadIdx.x & 31;                     // column group (8 cols)
    int tr = threadIdx.x >> 5;                     // row group    (8 rows)
    int c0 = tc * 8;

    float acc[8][8];
#pragma unroll
    for (int i = 0; i < 8; ++i)
#pragma unroll
        for (int j = 0; j < 8; ++j) acc[i][j] = 3.0e38f;

    for (int k = 0; k < NN; ++k) {
        float a[8];
#pragma unroll
        for (int i = 0; i < 8; ++i) a[i] = As[(tr * 8 + i) * NN + k]; // LDS broadcast
        const float4 b0 = *(const float4*)(Db + k * NN + c0);
        const float4 b1 = *(const float4*)(Db + k * NN + c0 + 4);
        float bv[8] = {b0.x, b0.y, b0.z, b0.w, b1.x, b1.y, b1.z, b1.w};
#pragma unroll
        for (int i = 0; i < 8; ++i)
#pragma unroll
            for (int j = 0; j < 8; ++j)
                acc[i][j] = fminf(acc[i][j], a[i] + bv[j]);
    }

    float* Ob = Dout + b * NN * NN;
#pragma unroll
    for (int i = 0; i < 8; ++i) {
        int r = rb + tr * 8 + i;
        float4 o0 = {acc[i][0], acc[i][1], acc[i][2], acc[i][3]};
        float4 o1 = {acc[i][4], acc[i][5], acc[i][6], acc[i][7]};
        *(float4*)(Ob + r * NN + c0)     = o0;
        *(float4*)(Ob + r * NN + c0 + 4) = o1;
    }
}

// ---------------------------------------------------------------------------
// K3: exact stable-argsort selection. One block per node.
// rank_j = #{k : (d_k,k) < (d_j,j)}; keep rank<32; ascending-index output.
// ---------------------------------------------------------------------------
__global__ void __launch_bounds__(256) k_select(const float* __restrict__ D,
                                                int* __restrict__ nbr) {
    int node = blockIdx.x;                         // b*NN + i
    const float* row = D + (long)node * NN;
    __shared__ float ds[NN];
    __shared__ unsigned char sel[NN];
    int j = threadIdx.x;
    ds[j] = row[j];
    __syncthreads();
    float dj = ds[j];
    int rank = 0;
    for (int k = 0; k < NN; ++k) {
        float dk = ds[k];
        rank += (dk < dj) || (dk == dj && k < j);
    }
    sel[j] = (rank < KNB) ? 1 : 0;
    __syncthreads();
    if (rank < KNB) {
        int pos = 0;
        for (int k = 0; k < j; ++k) pos += sel[k];
        nbr[node * KNB + pos] = j;
    }
}

// ---------------------------------------------------------------------------
// K4a: sub_f = onehot(nbr) @ f  via V_WMMA_F32_16X16X4_F32 (bit-exact: every
// accumulator sum has exactly one nonzero {0,1}-weighted term).
// Block = 8 waves per node. Each wave owns ONE 16-wide feature-column tile
// (n0 = wave*16) and keeps TWO accumulators (neighbor rows 0-15 and 16-31),
// so each B-fragment load feeds two WMMAs.
// ---------------------------------------------------------------------------
__global__ void __launch_bounds__(256) k_subf(const float* __restrict__ feat,
                                              const int* __restrict__ nbr,
                                              float* __restrict__ outF) {
    int node = blockIdx.x;                         // b*NN + i
    int b    = node >> 8;
    const float* fb = feat + (long)b * NN * FF;
    int wave  = threadIdx.x >> 5;
    int lane  = threadIdx.x & 31;
    int lhalf = lane >> 4;                         // K sub-pair select
    int lm    = lane & 15;                         // M (A) / N (B,D) index
    int n0    = wave * 16;                         // this wave's feature cols

    int nb0 = nbr[node * KNB + lm];                // neighbor slots 0..15
    int nb1 = nbr[node * KNB + 16 + lm];           // neighbor slots 16..31

    v8f c0 = {};
    v8f c1 = {};
    for (int k0 = 0; k0 < NN; k0 += 4) {
        int kk = k0 + lhalf * 2;
        __builtin_prefetch(fb + (k0 + 8) * FF + n0, 0, 3);
        v2f bf;                                    // shared B operand
        bf.x = fb[kk * FF + n0 + lm];
        bf.y = fb[(kk + 1) * FF + n0 + lm];
        v2f a0, a1;                                // one-hot A fragments
        a0.x = (nb0 == kk)     ? 1.0f : 0.0f;
        a0.y = (nb0 == kk + 1) ? 1.0f : 0.0f;
        a1.x = (nb1 == kk)     ? 1.0f : 0.0f;
        a1.y = (nb1 == kk + 1) ? 1.0f : 0.0f;
        c0 = __builtin_amdgcn_wmma_f32_16x16x4_f32(
                 false, a0, false, bf, (short)0, c0, false, false);
        c1 = __builtin_amdgcn_wmma_f32_16x16x4_f32(
                 false, a1, false, bf, (short)0, c1, false, false);
    }
    // D layout: VGPR v -> M=v (lanes 0-15) / M=v+8 (lanes 16-31), N=lm
    float* o = outF + (long)node * KNB * FF + n0;
#pragma unroll
    for (int v = 0; v < 8; ++v) {
        int m = v + lhalf * 8;
        o[m * FF + lm]          = c0[v];           // rows 0..15
        o[(16 + m) * FF + lm]   = c1[v];           // rows 16..31
    }
}

// ---------------------------------------------------------------------------
// K4b: direct gathers for sub_a (N,32,32) and sub_e (N,32,32,3)
// ---------------------------------------------------------------------------
__global__ void __launch_bounds__(256) k_subae(const float* __restrict__ adj,
                                               const float* __restrict__ ef,
                                               const int* __restrict__ nbr,
                                               float* __restrict__ outA,
                                               float* __restrict__ outE) {
    int node = blockIdx.x;
    int b    = node >> 8;
    __shared__ int nb[KNB];
    if (threadIdx.x < KNB) nb[threadIdx.x] = nbr[node * KNB + threadIdx.x];
    __syncthreads();
    const float* ab = adj + (long)b * NN * NN;
    const float* eb = ef  + (long)b * NN * NN * 3;
    for (int pq = threadIdx.x; pq < KNB * KNB; pq += 256) {
        int p = pq >> 5, q = pq & 31;
        int u = nb[p], v = nb[q];
        outA[(long)node * KNB * KNB + pq] = ab[u * NN + v];
        long eo = ((long)node * KNB * KNB + pq) * 3;
        const float* es = eb + ((long)u * NN + v) * 3;
        outE[eo + 0] = es[0];
        outE[eo + 1] = es[1];
        outE[eo + 2] = es[2];
    }
}

// ---------------------------------------------------------------------------
extern "C" void kernel_launch(void* const* d_in, const int* in_sizes, int n_in,
                              void* d_out, int out_size, void* d_ws, size_t ws_size,
                              hipStream_t stream) {
    const float* feat = (const float*)d_in[0];   // (16,256,128)
    const float* adj  = (const float*)d_in[1];   // (16,256,256)
    const float* ef   = (const float*)d_in[2];   // (16,256,256,3)
    // d_in[3] = neigh_size (32), fixed by the reference setup

    float* D0  = (float*)d_ws;                   // 4 MB
    float* D1  = D0 + (long)BATCH * NN * NN;     // 4 MB
    int*   nbr = (int*)(D1 + (long)BATCH * NN * NN); // 512 KB

    float* outF = (float*)d_out;
    float* outA = outF + (long)BATCH * NN * KNB * FF;
    float* outE = outA + (long)BATCH * NN * KNB * KNB;

    k_init<<<BATCH * NN * NN / 256, 256, 0, stream>>>(adj, D0);

    float* a = D0;
    float* bq = D1;
    for (int it = 0; it < 8; ++it) {             // ceil(log2(255)) = 8
        k_minplus<<<BATCH * 4, 256, 0, stream>>>(a, bq);
        float* t = a; a = bq; bq = t;
    }

    k_select<<<BATCH * NN, 256, 0, stream>>>(a, nbr);
    k_subf  <<<BATCH * NN, 256, 0, stream>>>(feat, nbr, outF);
    k_subae <<<BATCH * NN, 256, 0, stream>>>(adj, ef, nbr, outA, outE);
}